// Text_encoder_61950608278168
// MI455X (gfx1250) — compile-verified
//
#include <hip/hip_runtime.h>

// ---------------------------------------------------------------------------
// Problem constants (from the reference)
// ---------------------------------------------------------------------------
#define N_SENT 384   // N
#define D_MLP  768   // D
#define W1_LD  1536  // W1 row length (2*D)

typedef float v2f __attribute__((ext_vector_type(2)));
typedef float v8f __attribute__((ext_vector_type(8)));

// ---------------------------------------------------------------------------
// Phase 1: A[i,o] = sum_k P[i,k] * W1[o,k]       (+ b1[o] folded in)
//          B[i,o] = sum_k P[i,k] * W1[o,768+k]
// One wave computes one 16(i) x 16(o) tile for BOTH halves, sharing the
// P fragment (A-operand) between the two v_wmma_f32_16x16x4_f32 chains.
//
// Fragment layouts (CDNA5 ISA 7.12.2, wave32):
//   A 16x4 f32 : lane L(0..15): a.x=K0,a.y=K1 of row M=L ; L(16..31): K2,K3
//   B 4x16 f32 : lane L: b.x=B[kk,n], b.y=B[kk+1,n],  n=L&15, kk=2*(L>>4)
//   D 16x16 f32: lane L, vgpr v -> M = v + 8*(L>>4), N = L&15
// ---------------------------------------------------------------------------
__global__ __launch_bounds__(128)
void proj_wmma_f32_kernel(const float* __restrict__ P,
                          const float* __restrict__ W1,
                          const float* __restrict__ b1,
                          float* __restrict__ Abuf,
                          float* __restrict__ Bbuf) {
  const int lane = threadIdx.x & 31;
  const int wave = threadIdx.x >> 5;
  const int i0 = blockIdx.x * 16;                 // row tile (sentence i)
  const int o0 = (blockIdx.y * 4 + wave) * 16;    // col tile (hidden o)

  const int mn = lane & 15;            // A-frag row (M) and B-frag col (N)
  const int kk = (lane >> 4) << 1;     // K sub-offset: 0 (lanes 0-15) / 2

  const float* pa  = P  + (size_t)(i0 + mn) * D_MLP + kk;         // P[i0+mn, kk+..]
  const float* pb0 = W1 + (size_t)(o0 + mn) * W1_LD + kk;         // W1a[o0+mn, ..]
  const float* pb1 = pb0 + D_MLP;                                 // W1b[o0+mn, ..]

  v8f accA = {};
  v8f accB = {};
  for (int k = 0; k < D_MLP; k += 4) {
    v2f a   = *(const v2f*)(pa  + k);
    v2f b0  = *(const v2f*)(pb0 + k);
    v2f b1v = *(const v2f*)(pb1 + k);
    // (neg_a, A, neg_b, B, c_mod, C, reuse_a, reuse_b)
    accA = __builtin_amdgcn_wmma_f32_16x16x4_f32(false, a, false, b0,
                                                 (short)0, accA, false, false);
    accB = __builtin_amdgcn_wmma_f32_16x16x4_f32(false, a, false, b1v,
                                                 (short)0, accB, false, false);
  }

  const int mbase = (lane >> 4) << 3;
  const int o = o0 + mn;
  const float bias = b1[o];            // fold b1 into the A half only
#pragma unroll
  for (int v = 0; v < 8; ++v) {
    const int m = v + mbase;
    Abuf[(size_t)(i0 + m) * D_MLP + o] = accA[v] + bias;
    Bbuf[(size_t)(i0 + m) * D_MLP + o] = accB[v];
  }
}

// ---------------------------------------------------------------------------
// tanh on the fast path: native v_tanh_f32 if the builtin exists for gfx1250,
// else a v_exp_f32 + v_rcp_f32 formulation (sign-safe, exact at 0 and +-inf).
// ---------------------------------------------------------------------------
__device__ __forceinline__ float fast_tanh(float x) {
#if __has_builtin(__builtin_amdgcn_tanhf)
  return __builtin_amdgcn_tanhf(x);
#elif __has_builtin(__builtin_amdgcn_tanh_f32)
  return __builtin_amdgcn_tanh_f32(x);
#else
  const float ax = __builtin_fabsf(x);
  const float e  = __builtin_amdgcn_exp2f(ax * -2.8853900817779268f); // exp(-2ax)
  float tv = (1.0f - e) * __builtin_amdgcn_rcpf(1.0f + e);
  return __builtin_copysignf(tv, x);
#endif
}

// ---------------------------------------------------------------------------
// Phase 2: logits[i,j,c] = sum_d tanh(A[i,d] + B[j,d]) * W2[c,d] + b2[c]
// 16x16 (i,j) pairs per 256-thread block; A/B rows staged through LDS in two
// D-chunks of 384. Padded row stride 388 floats -> bank stride 4 -> the 16
// distinct jj rows in a wave hit 16 distinct bank groups (conflict-free).
// ---------------------------------------------------------------------------
#define D_CH 384
#define ROWP 388   // 384 + 4 pad; multiple of 4 keeps float4 alignment

__global__ __launch_bounds__(256)
void pair_tanh_logits_kernel(const float* __restrict__ Abuf,
                             const float* __restrict__ Bbuf,
                             const float* __restrict__ W2,
                             const float* __restrict__ b2,
                             float* __restrict__ out) {
  __shared__ float as[16 * ROWP];
  __shared__ float bs[16 * ROWP];

  const int i0 = blockIdx.x * 16;
  const int j0 = blockIdx.y * 16;
  const int t  = threadIdx.x;
  const int ii = t >> 4;
  const int jj = t & 15;

  const float* w0 = W2;            // W2[0,:]
  const float* w1 = W2 + D_MLP;    // W2[1,:]

  float acc0 = 0.f, acc1 = 0.f;

  for (int c0 = 0; c0 < D_MLP; c0 += D_CH) {
    // Cooperative stage: 16 rows x 384 floats of each of A,B as float4.
    for (int idx = t; idx < 16 * (D_CH / 4); idx += 256) {
      const int r  = idx / (D_CH / 4);
      const int c4 = (idx % (D_CH / 4)) * 4;
      *(float4*)&as[r * ROWP + c4] =
          *(const float4*)&Abuf[(size_t)(i0 + r) * D_MLP + c0 + c4];
      *(float4*)&bs[r * ROWP + c4] =
          *(const float4*)&Bbuf[(size_t)(j0 + r) * D_MLP + c0 + c4];
    }
    __syncthreads();

    const float* ar = &as[ii * ROWP];
    const float* br = &bs[jj * ROWP];
    for (int d = 0; d < D_CH; d += 4) {
      const float4 av  = *(const float4*)(ar + d);
      const float4 bv  = *(const float4*)(br + d);
      const float4 w0v = *(const float4*)(w0 + c0 + d);
      const float4 w1v = *(const float4*)(w1 + c0 + d);
#pragma unroll
      for (int q = 0; q < 4; ++q) {
        const float x  = (&av.x)[q] + (&bv.x)[q];
        const float tv = fast_tanh(x);
        acc0 = __builtin_fmaf(tv, (&w0v.x)[q], acc0);
        acc1 = __builtin_fmaf(tv, (&w1v.x)[q], acc1);
      }
    }
    __syncthreads();   // protect LDS before next chunk's stage
  }

  const size_t oidx = ((size_t)(i0 + ii) * N_SENT + (j0 + jj)) * 2;
  float2 r;
  r.x = acc0 + b2[0];
  r.y = acc1 + b2[1];
  *(float2*)&out[oidx] = r;
}

// ---------------------------------------------------------------------------
// Launcher
// Inputs (setup_inputs order): paramemory[N*D], W1[D*2D], b1[D], W2[C*D], b2[C]
// Output: logits [N, N, C] f32.  Workspace: Abuf + Bbuf = 2 * N * D floats.
// ---------------------------------------------------------------------------
extern "C" void kernel_launch(void* const* d_in, const int* in_sizes, int n_in,
                              void* d_out, int out_size, void* d_ws, size_t ws_size,
                              hipStream_t stream) {
  const float* P  = (const float*)d_in[0];
  const float* W1 = (const float*)d_in[1];
  const float* b1 = (const float*)d_in[2];
  const float* W2 = (const float*)d_in[3];
  const float* b2 = (const float*)d_in[4];
  float* out = (float*)d_out;

  float* Abuf = (float*)d_ws;
  float* Bbuf = Abuf + (size_t)N_SENT * D_MLP;

  // Phase 1: 24 i-tiles x 48 o-tiles; 4 waves (o-tiles) per 128-thread block.
  dim3 g1(N_SENT / 16, (D_MLP / 16) / 4);
  proj_wmma_f32_kernel<<<g1, 128, 0, stream>>>(P, W1, b1, Abuf, Bbuf);

  // Phase 2: 24 x 24 tiles of 16x16 (i,j) pairs.
  dim3 g2(N_SENT / 16, N_SENT / 16);
  pair_tanh_logits_kernel<<<g2, 256, 0, stream>>>(Abuf, Bbuf, W2, b2, out);
}